// Cross_modal_Center_ContrastiveLoss3_36618891166027
// MI455X (gfx1250) — compile-verified
//
#include <hip/hip_runtime.h>

#define N_SAMP 4096
#define DIM    512
#define NCLS   395
#define CPAD   400
#define NT     25      // 16x16 tiles per side (400/16)
#define MARGIN 0.5f

typedef __attribute__((ext_vector_type(2))) float v2f;
typedef __attribute__((ext_vector_type(8))) float v8f;

__device__ __forceinline__ void fatomic_add(float* p, float v) {
    __hip_atomic_fetch_add(p, v, __ATOMIC_RELAXED, __HIP_MEMORY_SCOPE_AGENT);
}

// K0: zero sums/counts, build zero-padded centers copy
__global__ void k_init(const float* __restrict__ centers,
                       float* __restrict__ m1, float* __restrict__ m2,
                       float* __restrict__ cen, float* __restrict__ counts) {
    int idx = blockIdx.x * blockDim.x + threadIdx.x;
    if (idx < CPAD * DIM) {
        int c = idx / DIM;
        m1[idx] = 0.0f;
        m2[idx] = 0.0f;
        cen[idx] = (c < NCLS) ? centers[idx] : 0.0f;
    }
    if (idx < CPAD) counts[idx] = 0.0f;
}

// K1: segment sums via hardware f32 global atomics (one block per sample)
__global__ void k_accum(const float* __restrict__ a1, const float* __restrict__ a2,
                        const int* __restrict__ tgt,
                        float* __restrict__ m1, float* __restrict__ m2,
                        float* __restrict__ counts) {
    int i = blockIdx.x;
    int t = tgt[i];
    const float* r1 = a1 + (size_t)i * DIM;
    const float* r2 = a2 + (size_t)i * DIM;
    float* o1 = m1 + (size_t)t * DIM;
    float* o2 = m2 + (size_t)t * DIM;
    for (int d = threadIdx.x; d < DIM; d += blockDim.x) {
        fatomic_add(o1 + d, r1[d]);
        fatomic_add(o2 + d, r2[d]);
    }
    if (threadIdx.x == 0) fatomic_add(counts + t, 1.0f);
}

// K2: means = sums / max(cnt,1) in place; row squared-norms for means1/means2/centers
__global__ void k_finalize(float* __restrict__ m1, float* __restrict__ m2,
                           const float* __restrict__ cen, const float* __restrict__ counts,
                           float* __restrict__ n1, float* __restrict__ n2,
                           float* __restrict__ ncn) {
    __shared__ float s1[256], s2[256], s3[256];
    int c = blockIdx.x;
    int tid = threadIdx.x;
    float inv = 1.0f / fmaxf(counts[c], 1.0f);
    float a = 0.0f, b = 0.0f, g = 0.0f;
    for (int d = tid; d < DIM; d += blockDim.x) {
        size_t o = (size_t)c * DIM + d;
        float v1 = m1[o] * inv; m1[o] = v1; a += v1 * v1;
        float v2 = m2[o] * inv; m2[o] = v2; b += v2 * v2;
        float vc = cen[o];                 g += vc * vc;
    }
    s1[tid] = a; s2[tid] = b; s3[tid] = g;
    __syncthreads();
    for (int off = 128; off > 0; off >>= 1) {
        if (tid < off) { s1[tid] += s1[tid + off]; s2[tid] += s2[tid + off]; s3[tid] += s3[tid + off]; }
        __syncthreads();
    }
    if (tid == 0) { n1[c] = s1[0]; n2[c] = s2[0]; ncn[c] = s3[0]; }
}

// K3: one wave per 16x16 tile of the CxC matrix; f32 WMMA over K=512,
// software-pipelined so next step's loads overlap current WMMAs.
__global__ __launch_bounds__(32)
void k_tile(const float* __restrict__ m1, const float* __restrict__ m2,
            const float* __restrict__ cen, const float* __restrict__ counts,
            const float* __restrict__ n1, const float* __restrict__ n2,
            const float* __restrict__ ncn, float* __restrict__ partial) {
    int tr = blockIdx.y, tc = blockIdx.x;
    int lane = threadIdx.x;
    int half = lane >> 4;          // 0 or 1
    int l16  = lane & 15;
    int row  = tr * 16 + l16;      // A-matrix row held by this lane
    int col  = tc * 16 + l16;      // B-matrix column held by this lane
    int koff = half * 2;           // K sub-offset per ISA 32-bit 16x4 layout

    const float* pa1 = m1  + (size_t)row * DIM + koff;
    const float* pa2 = m2  + (size_t)row * DIM + koff;
    const float* pb  = cen + (size_t)col * DIM + koff;

    v8f acc1 = {0.f,0.f,0.f,0.f,0.f,0.f,0.f,0.f};
    v8f acc2 = {0.f,0.f,0.f,0.f,0.f,0.f,0.f,0.f};

    // Stage 0 prefetch
    v2f A1 = *(const v2f*)(pa1);
    v2f A2 = *(const v2f*)(pa2);
    v2f B  = *(const v2f*)(pb);

#pragma unroll 4
    for (int k = 0; k < DIM - 4; k += 4) {
        // Issue next step's loads before consuming current registers:
        // lets the compiler wait with loadcnt<=3 instead of 0.
        v2f nA1 = *(const v2f*)(pa1 + k + 4);
        v2f nA2 = *(const v2f*)(pa2 + k + 4);
        v2f nB  = *(const v2f*)(pb  + k + 4);
        acc1 = __builtin_amdgcn_wmma_f32_16x16x4_f32(false, A1, false, B, (short)0, acc1, false, false);
        acc2 = __builtin_amdgcn_wmma_f32_16x16x4_f32(false, A2, false, B, (short)0, acc2, false, false);
        A1 = nA1; A2 = nA2; B = nB;
    }
    acc1 = __builtin_amdgcn_wmma_f32_16x16x4_f32(false, A1, false, B, (short)0, acc1, false, false);
    acc2 = __builtin_amdgcn_wmma_f32_16x16x4_f32(false, A2, false, B, (short)0, acc2, false, false);

    // Epilogue: lane holds D[m][n] for n = col, m = tr*16 + r + half*8 (r = VGPR index)
    float cntB = counts[col];
    float nrmB = ncn[col];
    float local = 0.0f;
#pragma unroll
    for (int r = 0; r < 8; ++r) {
        int m = tr * 16 + r + half * 8;
        float w = counts[m] * cntB;
        float sq1 = n1[m] + nrmB - 2.0f * acc1[r];
        float sq2 = n2[m] + nrmB - 2.0f * acc2[r];
        float d1 = sqrtf(fmaxf(sq1, 1e-12f));
        float d2 = sqrtf(fmaxf(sq2, 1e-12f));
        float neg1 = fmaxf(MARGIN - d1, 0.0f);
        float neg2 = fmaxf(MARGIN - d2, 0.0f);
        bool diag = (m == col);
        float f1 = diag ? d1 * d1 : neg1 * neg1;
        float f2 = diag ? d2 * d2 : neg2 * neg2;
        local += w * (f1 + f2);
    }
    for (int off = 16; off > 0; off >>= 1)
        local += __shfl_xor(local, off, 32);
    if (lane == 0) partial[tr * NT + tc] = local;
}

// K4: reduce 625 tile partials, scale by 1/N^2
__global__ void k_reduce(const float* __restrict__ partial, float* __restrict__ out) {
    __shared__ float s[256];
    int tid = threadIdx.x;
    float v = 0.0f;
    for (int i = tid; i < NT * NT; i += 256) v += partial[i];
    s[tid] = v;
    __syncthreads();
    for (int off = 128; off > 0; off >>= 1) {
        if (tid < off) s[tid] += s[tid + off];
        __syncthreads();
    }
    if (tid == 0) out[0] = s[0] * (1.0f / ((float)N_SAMP * (float)N_SAMP));
}

extern "C" void kernel_launch(void* const* d_in, const int* in_sizes, int n_in,
                              void* d_out, int out_size, void* d_ws, size_t ws_size,
                              hipStream_t stream) {
    const float* modal1  = (const float*)d_in[0];
    const float* modal2  = (const float*)d_in[1];
    const float* centers = (const float*)d_in[2];
    const int*   targets = (const int*)d_in[3];

    float* ws     = (float*)d_ws;
    float* m1     = ws;                    // CPAD*DIM  (sums -> means, modal1)
    float* m2     = m1  + CPAD * DIM;      // CPAD*DIM  (sums -> means, modal2)
    float* cen    = m2  + CPAD * DIM;      // CPAD*DIM  (zero-padded centers)
    float* counts = cen + CPAD * DIM;      // CPAD
    float* n1     = counts + CPAD;         // CPAD
    float* n2     = n1  + CPAD;            // CPAD
    float* ncn    = n2  + CPAD;            // CPAD
    float* part   = ncn + CPAD;            // NT*NT

    k_init<<<(CPAD * DIM + 255) / 256, 256, 0, stream>>>(centers, m1, m2, cen, counts);
    k_accum<<<N_SAMP, 256, 0, stream>>>(modal1, modal2, targets, m1, m2, counts);
    k_finalize<<<CPAD, 256, 0, stream>>>(m1, m2, cen, counts, n1, n2, ncn);
    dim3 grid(NT, NT);
    k_tile<<<grid, 32, 0, stream>>>(m1, m2, cen, counts, n1, n2, ncn, part);
    k_reduce<<<1, 256, 0, stream>>>(part, (float*)d_out);
}